// mRNN_72301479461003
// MI455X (gfx1250) — compile-verified
//
#include <hip/hip_runtime.h>

typedef float v2f __attribute__((ext_vector_type(2)));
typedef float v8f __attribute__((ext_vector_type(8)));
typedef unsigned int v4u __attribute__((ext_vector_type(4)));
typedef int v4i __attribute__((ext_vector_type(4)));
typedef int v8i __attribute__((ext_vector_type(8)));

#define RN 2048      // hidden size N
#define RB 64        // batch B
#define RTT 512      // timesteps T
#define RI 128       // input size I
#define STRIP 32     // N-columns owned by one workgroup (LDS-resident weights)
#define NWG (RN / STRIP)   // 64 persistent workgroups
#define ALPHA 0.1f

#if defined(__has_builtin)
#if __has_builtin(__builtin_amdgcn_tensor_load_to_lds)
#define HAVE_TDM 1
#endif
#endif

// ---------------------------------------------------------------------------
// All transposed operands are stored K-PAIR-INTERLEAVED: element (k, c) lives
// at [(k>>1)*2*C + c*2 + (k&1)], so the fp32-WMMA fragment pair (k, k+1) for a
// lane is one aligned 8-byte load (global_load_b64 / ds_load_b64).
// ---------------------------------------------------------------------------

// W_effTp[(k>>1)*2N + n*2 + (k&1)] = relu(W_rec[n][k]) * mask[n][k] * sign[n][k]
__global__ void prep_weffT(const float* __restrict__ Wrec, const int* __restrict__ Wmask,
                           const int* __restrict__ Wsign, float* __restrict__ WeffTp) {
  __shared__ float tile[32][33];
  const int bn = blockIdx.x, bk = blockIdx.y;
  const int col = threadIdx.x & 31, row = threadIdx.x >> 5;
#pragma unroll
  for (int j = 0; j < 32; j += 8) {
    const int n = bn * 32 + row + j;
    const int k = bk * 32 + col;
    float w = Wrec[n * RN + k];
    w = w > 0.f ? w : 0.f;
    const int ms = Wmask[n * RN + k] * Wsign[n * RN + k];
    tile[row + j][col] = w * (float)ms;
  }
  __syncthreads();
#pragma unroll
  for (int j = 0; j < 32; j += 8) {
    const int k = bk * 32 + row + j;
    const int n = bn * 32 + col;
    WeffTp[(size_t)(k >> 1) * (2 * RN) + n * 2 + (k & 1)] = tile[col][row + j];
  }
}

// W_inpTp[(i>>1)*2N + n*2 + (i&1)] = W_inp[n][i]
__global__ void prep_winpT(const float* __restrict__ Winp, float* __restrict__ WinpTp) {
  __shared__ float tile[32][33];
  const int bn = blockIdx.x, bi = blockIdx.y;
  const int col = threadIdx.x & 31, row = threadIdx.x >> 5;
#pragma unroll
  for (int j = 0; j < 32; j += 8)
    tile[row + j][col] = Winp[(bn * 32 + row + j) * RI + bi * 32 + col];
  __syncthreads();
#pragma unroll
  for (int j = 0; j < 32; j += 8) {
    const int i = bi * 32 + row + j;
    const int n = bn * 32 + col;
    WinpTp[(size_t)(i >> 1) * (2 * RN) + n * 2 + (i & 1)] = tile[col][row + j];
  }
}

// xTp[t*I*B + (i>>1)*2B + b*2 + (i&1)] = x[b][t][i]
__global__ void prep_xT(const float* __restrict__ x, float* __restrict__ xTp) {
  __shared__ float tile[32][33];
  const int bb = blockIdx.x, bt = blockIdx.y;   // b-tile, (t*I+i)-tile
  const int col = threadIdx.x & 31, row = threadIdx.x >> 5;
#pragma unroll
  for (int j = 0; j < 32; j += 8)
    tile[row + j][col] = x[(size_t)(bb * 32 + row + j) * (RTT * RI) + bt * 32 + col];
  __syncthreads();
#pragma unroll
  for (int j = 0; j < 32; j += 8) {
    const int ti = bt * 32 + row + j;
    const int t = ti >> 7, i = ti & (RI - 1);
    const int b = bb * 32 + col;
    xTp[(size_t)t * (RI * RB) + (i >> 1) * (2 * RB) + b * 2 + (i & 1)] = tile[col][row + j];
  }
}

// h = h0 copy;  rT0p[(n>>1)*2B + b*2 + (n&1)] = relu(h0[b][n])
__global__ void prep_h(const float* __restrict__ h0, float* __restrict__ h,
                       float* __restrict__ rT0p) {
  __shared__ float tile[32][33];
  const int bb = blockIdx.x, bn = blockIdx.y;
  const int col = threadIdx.x & 31, row = threadIdx.x >> 5;
#pragma unroll
  for (int j = 0; j < 32; j += 8) {
    const int idx = (bb * 32 + row + j) * RN + bn * 32 + col;
    const float v = h0[idx];
    h[idx] = v;
    tile[row + j][col] = v;
  }
  __syncthreads();
#pragma unroll
  for (int j = 0; j < 32; j += 8) {
    const int n = bn * 32 + row + j;
    const int b = bb * 32 + col;
    const float v = tile[col][row + j];
    rT0p[(n >> 1) * (2 * RB) + b * 2 + (n & 1)] = v > 0.f ? v : 0.f;
  }
}

__global__ void init_sync(int* __restrict__ s) {
  if (threadIdx.x < 2) s[threadIdx.x] = 0;
}

// ---------------------------------------------------------------------------
// TDM 2D tile load: global (strided rows) -> LDS (contiguous). D# built per
// CDNA5 ISA 8.3/8.4: group0 = {count/type/addrs}, group1 = {data_size, dims,
// tile dims, dim0 stride}. 2D tensor -> groups 2/3 disabled (zero).
// ---------------------------------------------------------------------------
#ifdef HAVE_TDM
__device__ static inline void tdm_load_2d(unsigned int lds_addr, const void* gaddr,
                                          unsigned int tile_d0, unsigned int tile_d1,
                                          unsigned int tensor_d0, unsigned int tensor_d1,
                                          unsigned int stride0_elems) {
  const unsigned long long ga = (unsigned long long)gaddr;
  v4u g0;
  g0.x = 1u;                                   // count=1, user descriptor, no gather
  g0.y = lds_addr;                             // lds_addr (bytes, wave-relative)
  g0.z = (unsigned int)ga;                     // global_addr[31:0]
  g0.w = (unsigned int)(ga >> 32) | (2u << 30);// global_addr[56:32] | type=2
  v8i g1;
  g1[0] = (int)(2u << 16);                     // workgroup_mask=0, data_size=2 (4B)
  g1[1] = (int)(tensor_d0 << 16);              // [63:48] = tensor_dim0[15:0]
  g1[2] = (int)((tensor_d0 >> 16) | (tensor_d1 << 16)); // dim0 hi | dim1 lo
  g1[3] = (int)((tensor_d1 >> 16) | (tile_d0 << 16));   // dim1 hi | tile_dim0
  g1[4] = (int)tile_d1;                        // tile_dim1 (tile_dim2 = 0)
  g1[5] = (int)stride0_elems;                  // tensor_dim0_stride[31:0]
  g1[6] = 0;                                   // stride0 hi | dim1_stride lo
  g1[7] = 0;
  v4i z4 = {0, 0, 0, 0};
#if __clang_major__ >= 23
  v8i z8 = {0, 0, 0, 0, 0, 0, 0, 0};
  __builtin_amdgcn_tensor_load_to_lds(g0, g1, z4, z4, z8, 0);
#else
  __builtin_amdgcn_tensor_load_to_lds(g0, g1, z4, z4, 0);
#endif
}
#endif

// ---------------------------------------------------------------------------
// Persistent RNN kernel: 64 WGs x 256 threads (8 wave32). WG wg owns output
// columns [wg*32, wg*32+32); its W_eff strip (256 KB) + W_inp strip (16 KB)
// are TDM-loaded into LDS once and reused for all 512 steps.
// ---------------------------------------------------------------------------
__launch_bounds__(256, 1)
__global__ void rnn_persist(const float* __restrict__ xTp, const float* __restrict__ WeffTp,
                            const float* __restrict__ WinpTp, const float* __restrict__ tonic,
                            float* __restrict__ h, float* __restrict__ rT0,
                            float* __restrict__ rT1, float* __restrict__ out,
                            int* __restrict__ syncw) {
  __shared__ float lds_w[RN * STRIP];    // 256 KB: pair-interleaved [kp][64]
  __shared__ float lds_wi[RI * STRIP];   //  16 KB: pair-interleaved [ip][64]

  const int wg = blockIdx.x;
  const int n0 = wg * STRIP;
  const int tid = threadIdx.x;

#ifdef HAVE_TDM
  if (tid < 32) {  // one wave issues the two DMA descriptors
    tdm_load_2d((unsigned int)(unsigned long long)&lds_w[0], WeffTp + n0 * 2,
                /*tile*/ 2 * STRIP, RN / 2, /*tensor*/ 2 * RN, RN / 2, 2 * RN);
    tdm_load_2d((unsigned int)(unsigned long long)&lds_wi[0], WinpTp + n0 * 2,
                /*tile*/ 2 * STRIP, RI / 2, /*tensor*/ 2 * RN, RI / 2, 2 * RN);
    __builtin_amdgcn_s_wait_tensorcnt(0);
  }
#else
  for (int idx = tid; idx < RN * STRIP; idx += 256) {
    const int kp = idx >> 6, c = idx & 63;
    lds_w[idx] = WeffTp[(size_t)kp * (2 * RN) + n0 * 2 + c];
  }
  for (int idx = tid; idx < RI * STRIP; idx += 256) {
    const int kp = idx >> 6, c = idx & 63;
    lds_wi[idx] = WinpTp[(size_t)kp * (2 * RN) + n0 * 2 + c];
  }
#endif
  __syncthreads();

  const int lane = tid & 31;
  const int wid  = tid >> 5;
  const int m0   = (wid >> 1) * 16;       // 0,16,32,48
  const int nn0  = (wid & 1) * 16;        // 0,16 within strip
  const int ll   = lane & 15;
  const int kh   = lane >> 4;             // K-half select for fp32 A/B fragments
  const int nglob = n0 + nn0 + ll;
  const float ton = tonic[nglob];

  // v2f fragment base pointers (kh folded in; loop offsets are uniform).
  const v2f* bK2  = ((const v2f*)lds_w)  + kh * (STRIP) + nn0 + ll;  // [kp][col]
  const v2f* wiK2 = ((const v2f*)lds_wi) + kh * (STRIP) + nn0 + ll;
  const int  rbase = kh * RB + m0 + ll;   // v2f index into [kp][b] panels
  const int  npair = (nglob >> 1) * (2 * RB) + (nglob & 1);  // scatter base for rT

  for (int t = 0; t < RTT; ++t) {
    const float* rTc = (t & 1) ? rT1 : rT0;   // read
    float*       rTn = (t & 1) ? rT0 : rT1;   // write

    v8f acc = {};

    // --- recurrent GEMM: acc += relu(h_t) @ W_eff^T, K = 2048 ---
    const v2f* aK2 = ((const v2f*)rTc) + rbase;
#pragma unroll 4
    for (int kk = 0; kk < RN; kk += 4) {
      v2f a = aK2[kk * (RB / 2)];     // global_load_b64
      v2f b = bK2[kk * (STRIP / 2)];  // ds_load_b64
      acc = __builtin_amdgcn_wmma_f32_16x16x4_f32(false, a, false, b,
                                                  (short)0, acc, false, false);
    }

    // --- input GEMM: acc += x_t @ W_inp^T, K = 128 (fused drive) ---
    const v2f* xK2 = ((const v2f*)xTp) + (size_t)t * (RI * RB / 2) + rbase;
#pragma unroll 4
    for (int kk = 0; kk < RI; kk += 4) {
      v2f a = xK2[kk * (RB / 2)];
      v2f b = wiK2[kk * (STRIP / 2)];
      acc = __builtin_amdgcn_wmma_f32_16x16x4_f32(false, a, false, b,
                                                  (short)0, acc, false, false);
    }

    // --- epilogue: leaky integration, outputs, pair-interleaved rates ---
#pragma unroll
    for (int g = 0; g < 8; ++g) {
      const int m = m0 + kh * 8 + g;                  // C layout: lanes16-31 hold M+8
      const size_t hidx = (size_t)m * RN + nglob;
      const float hold = h[hidx];
      const float hnew = (1.0f - ALPHA) * hold + ALPHA * (acc[g] + ton);
      const float r = hnew > 0.f ? hnew : 0.f;
      h[hidx] = hnew;
      __builtin_nontemporal_store(r, &out[((size_t)m * RTT + t) * RN + nglob]);
      rTn[npair + m * 2] = r;
    }

    // --- device-wide barrier between steps ---
    if (t + 1 < RTT) {
      __builtin_amdgcn_fence(__ATOMIC_RELEASE, "agent");   // flush rT stores (global_wb)
      __syncthreads();
      if (tid == 0) {
        const int target = (t + 1) * NWG;
        const int prev = __hip_atomic_fetch_add(&syncw[0], 1, __ATOMIC_RELAXED,
                                                __HIP_MEMORY_SCOPE_AGENT);
        if (prev == target - 1) {
          __hip_atomic_store(&syncw[1], t + 1, __ATOMIC_RELEASE,
                             __HIP_MEMORY_SCOPE_AGENT);
        } else {
          while (__hip_atomic_load(&syncw[1], __ATOMIC_RELAXED,
                                   __HIP_MEMORY_SCOPE_AGENT) < t + 1) {
            __builtin_amdgcn_s_sleep(2);
          }
        }
      }
      __syncthreads();
      __builtin_amdgcn_fence(__ATOMIC_ACQUIRE, "agent");   // invalidate L0 (global_inv)
    }
  }
}

// ---------------------------------------------------------------------------
extern "C" void kernel_launch(void* const* d_in, const int* in_sizes, int n_in,
                              void* d_out, int out_size, void* d_ws, size_t ws_size,
                              hipStream_t stream) {
  const float* x     = (const float*)d_in[0];   // [B,T,I]
  const float* h0    = (const float*)d_in[1];   // [B,N]
  const float* Wrec  = (const float*)d_in[2];   // [N,N]
  const float* Winp  = (const float*)d_in[3];   // [N,I]
  const float* tonic = (const float*)d_in[4];   // [N]
  const int*   Wmask = (const int*)d_in[5];     // [N,N]
  const int*   Wsign = (const int*)d_in[6];     // [N,N]
  float* out = (float*)d_out;                   // [B,T,N]

  float* ws     = (float*)d_ws;                 // ~36.2 MB used
  float* WeffTp = ws;                           // [N/2][2N]
  float* WinpTp = WeffTp + (size_t)RN * RN;     // [I/2][2N]
  float* xTp    = WinpTp + (size_t)RI * RN;     // [T][I/2][2B]
  float* h      = xTp + (size_t)RTT * RI * RB;  // [B][N]
  float* rT0    = h + (size_t)RB * RN;          // [N/2][2B]
  float* rT1    = rT0 + (size_t)RN * RB;        // [N/2][2B]
  int*   syncw  = (int*)(rT1 + (size_t)RN * RB);// 2 ints

  prep_weffT<<<dim3(RN / 32, RN / 32), 256, 0, stream>>>(Wrec, Wmask, Wsign, WeffTp);
  prep_winpT<<<dim3(RN / 32, RI / 32), 256, 0, stream>>>(Winp, WinpTp);
  prep_xT<<<dim3(RB / 32, (RTT * RI) / 32), 256, 0, stream>>>(x, xTp);
  prep_h<<<dim3(RB / 32, RN / 32), 256, 0, stream>>>(h0, h, rT0);
  init_sync<<<1, 32, 0, stream>>>(syncw);
  rnn_persist<<<NWG, 256, 0, stream>>>(xTp, WeffTp, WinpTp, tonic, h, rT0, rT1, out, syncw);
}